// LSTM_18107582120063
// MI455X (gfx1250) — compile-verified
//
#include <hip/hip_runtime.h>
#include <hip/hip_bf16.h>

// Problem dims (match reference)
#define SEQN 512
#define BN   256
#define INN  256
#define HN   512
#define OUTN 128
#define KTOT (INN + HN)
#define KC   64          // K-chunk staged in LDS (divides INN and KTOT)

typedef __attribute__((ext_vector_type(2))) float v2f;
typedef __attribute__((ext_vector_type(8))) float v8f;

__device__ __forceinline__ float sigmoid_(float x) {
    return 1.0f / (1.0f + __expf(-x));
}

// ---------------------------------------------------------------------------
// Zero-init workspace region (h/c ping-pong buffers).
// ---------------------------------------------------------------------------
__global__ __launch_bounds__(256) void lstm_init_kernel(float* __restrict__ p, int n) {
    int i = blockIdx.x * blockDim.x + threadIdx.x;
    if (i < n) p[i] = 0.0f;
}

// ---------------------------------------------------------------------------
// One LSTM time step.
// Block = 256 threads = 8 wave32s. Block owns one 16-wide N (hidden) tile and
// 8 M (batch) tiles; each wave computes one 16x16 tile with 4 fp32 WMMA
// accumulators (f, i, c-cand, o). Gate weights for the current K-chunk are
// cooperatively staged in LDS (16 KB) and shared by all 8 waves, cutting
// global weight traffic 8x on the serial recurrence path.
//
// WMMA fp32 operand layout (V_WMMA_F32_16X16X4_F32):
//   A (16x4): lane<16 holds A[m0+lane][k], A[..][k+1]; lane>=16 holds k+2,k+3
//   B (4x16): lane<16 holds B[k][n0+lane], B[k+1][..]; lane>=16 holds k+2,k+3
//   C/D     : VGPR r: lane<16 -> (m0+r, n0+lane); lane>=16 -> (m0+8+r, ...)
// ---------------------------------------------------------------------------
__global__ __launch_bounds__(256) void lstm_step_kernel(
    const float* __restrict__ x_t,                       // [BN, INN]
    const float* __restrict__ Wf, const float* __restrict__ bf,
    const float* __restrict__ Wi, const float* __restrict__ bi,
    const float* __restrict__ Wc, const float* __restrict__ bc,
    const float* __restrict__ Wo, const float* __restrict__ bo,  // W: [KTOT, HN]
    const float* __restrict__ h_prev, const float* __restrict__ c_prev, // [BN, HN]
    float* __restrict__ h_next, float* __restrict__ c_next)             // [BN, HN]
{
    __shared__ float ldsB[4][KC][16];          // 4 gates x 64 K x 16 N = 16 KB

    const int tid  = threadIdx.x;
    const int lane = tid & 31;
    const int wib  = tid >> 5;                 // wave-in-block: 0..7
    const int lrow = lane & 15;
    const int lhi  = lane >> 4;                // 0 or 1

    const int nt = blockIdx.x & 31;            // HN/16 = 32 n-tiles
    const int mg = blockIdx.x >> 5;            // 2 m-groups of 8 waves
    const int m0 = (mg * 8 + wib) << 4;        // this wave's 16 batch rows
    const int n0 = nt << 4;
    const int col = n0 + lrow;                 // output / B column for this lane

    const float* arow_x = x_t    + (size_t)(m0 + lrow) * INN;
    const float* arow_h = h_prev + (size_t)(m0 + lrow) * HN;

    v8f accF = {}; v8f accI = {}; v8f accC = {}; v8f accO = {};

    for (int kc = 0; kc < KTOT; kc += KC) {
        __syncthreads();
        // ---- cooperative stage of the 4 gate weight chunks into LDS ----
        // KC*16 = 1024 elements per gate; 256 threads -> 4 per gate per thread
        #pragma unroll
        for (int j = 0; j < 4; ++j) {
            const int e  = tid + 256 * j;      // 0..1023
            const int kr = e >> 4;             // 0..63
            const int nn = e & 15;
            const size_t g = (size_t)(kc + kr) * HN + n0 + nn;
            ldsB[0][kr][nn] = Wf[g];
            ldsB[1][kr][nn] = Wi[g];
            ldsB[2][kr][nn] = Wc[g];
            ldsB[3][kr][nn] = Wo[g];
        }
        __syncthreads();

        // chunk is homogeneous (x part for kc<INN, h part otherwise)
        const float* arow = (kc < INN) ? (arow_x + kc) : (arow_h + (kc - INN));

        #pragma unroll 4
        for (int k = 0; k < KC; k += 4) {
            const int kl = k + 2 * lhi;
            v2f a;  a.x  = arow[kl];           a.y  = arow[kl + 1];
            v2f bF; bF.x = ldsB[0][kl][lrow];  bF.y = ldsB[0][kl + 1][lrow];
            v2f bI; bI.x = ldsB[1][kl][lrow];  bI.y = ldsB[1][kl + 1][lrow];
            v2f bC; bC.x = ldsB[2][kl][lrow];  bC.y = ldsB[2][kl + 1][lrow];
            v2f bO; bO.x = ldsB[3][kl][lrow];  bO.y = ldsB[3][kl + 1][lrow];

            accF = __builtin_amdgcn_wmma_f32_16x16x4_f32(false, a, false, bF, (short)0, accF, false, false);
            accI = __builtin_amdgcn_wmma_f32_16x16x4_f32(false, a, false, bI, (short)0, accI, false, false);
            accC = __builtin_amdgcn_wmma_f32_16x16x4_f32(false, a, false, bC, (short)0, accC, false, false);
            accO = __builtin_amdgcn_wmma_f32_16x16x4_f32(false, a, false, bO, (short)0, accO, false, false);
        }
    }

    // ---- fused gate nonlinearities + cell/hidden update ----
    const float bfv = bf[col];
    const float biv = bi[col];
    const float bcv = bc[col];
    const float bov = bo[col];

    #pragma unroll
    for (int r = 0; r < 8; ++r) {
        const int row = m0 + r + 8 * lhi;
        const size_t idx = (size_t)row * HN + col;
        const float fg   = sigmoid_(accF[r] + bfv);
        const float ig   = sigmoid_(accI[r] + biv);
        const float cand = __tanf(accC[r] + bcv);     // faithful: tan, not tanh
        const float og   = sigmoid_(accO[r] + bov);
        const float cn   = fg * c_prev[idx] + ig * cand;
        c_next[idx] = cn;
        h_next[idx] = og * __tanf(cn);                // faithful: tan, not tanh
    }
}

// ---------------------------------------------------------------------------
// logits = h_last @ Wout + bout  via fp32 WMMA. 128 waves (16x8 tiles).
// ---------------------------------------------------------------------------
__global__ __launch_bounds__(256) void out_gemm_kernel(
    const float* __restrict__ h_last,    // [BN, HN]
    const float* __restrict__ Wout,      // [HN, OUTN]
    const float* __restrict__ bout,      // [OUTN]
    float* __restrict__ logits)          // [BN, OUTN]
{
    const int gid  = blockIdx.x * blockDim.x + threadIdx.x;
    const int wave = gid >> 5;                 // 128 waves total
    const int lane = threadIdx.x & 31;
    const int lrow = lane & 15;
    const int lhi  = lane >> 4;

    const int nt = wave & 7;                   // OUTN/16 = 8
    const int mt = wave >> 3;                  // BN/16 = 16
    const int m0 = mt << 4;
    const int n0 = nt << 4;
    const int col = n0 + lrow;

    const float* arow = h_last + (size_t)(m0 + lrow) * HN;

    v8f acc = {};
    #pragma unroll 4
    for (int k = 0; k < HN; k += 4) {
        const int ka = k + 2 * lhi;
        v2f a; a.x = arow[ka]; a.y = arow[ka + 1];
        v2f b; b.x = Wout[(size_t)ka * OUTN + col]; b.y = Wout[(size_t)(ka + 1) * OUTN + col];
        acc = __builtin_amdgcn_wmma_f32_16x16x4_f32(false, a, false, b, (short)0, acc, false, false);
    }

    const float bb = bout[col];
    #pragma unroll
    for (int r = 0; r < 8; ++r) {
        const int row = m0 + r + 8 * lhi;
        logits[(size_t)row * OUTN + col] = acc[r] + bb;
    }
}

// ---------------------------------------------------------------------------
// Row-wise log-softmax over OUTN=128 columns. One wave32 per row; each lane
// owns 4 columns; reductions via __shfl_xor (width 32 — wave32 only on CDNA5).
// ---------------------------------------------------------------------------
__global__ __launch_bounds__(256) void logsoftmax_kernel(
    const float* __restrict__ logits,    // [BN, OUTN]
    float* __restrict__ out)             // [BN, OUTN]
{
    const int gid  = blockIdx.x * blockDim.x + threadIdx.x;
    const int wave = gid >> 5;                 // one wave per row
    const int lane = threadIdx.x & 31;
    if (wave >= BN) return;

    const float* rowp = logits + (size_t)wave * OUTN;
    float v[4];
    float m = -__builtin_inff();
    #pragma unroll
    for (int j = 0; j < 4; ++j) {
        v[j] = rowp[lane + 32 * j];
        m = fmaxf(m, v[j]);
    }
    #pragma unroll
    for (int off = 16; off >= 1; off >>= 1)
        m = fmaxf(m, __shfl_xor(m, off, 32));

    float s = 0.0f;
    #pragma unroll
    for (int j = 0; j < 4; ++j) s += __expf(v[j] - m);
    #pragma unroll
    for (int off = 16; off >= 1; off >>= 1)
        s += __shfl_xor(s, off, 32);

    const float ls = __logf(s);
    #pragma unroll
    for (int j = 0; j < 4; ++j)
        out[(size_t)wave * OUTN + lane + 32 * j] = v[j] - m - ls;
}

// ---------------------------------------------------------------------------
// Host-side launcher (graph-capture safe: only kernel launches on `stream`).
// ---------------------------------------------------------------------------
extern "C" void kernel_launch(void* const* d_in, const int* in_sizes, int n_in,
                              void* d_out, int out_size, void* d_ws, size_t ws_size,
                              hipStream_t stream) {
    const float* batch = (const float*)d_in[0];   // [SEQ, BN, INN]
    const float* Wf    = (const float*)d_in[1];
    const float* bf    = (const float*)d_in[2];
    const float* Wi    = (const float*)d_in[3];
    const float* bi    = (const float*)d_in[4];
    const float* Wc    = (const float*)d_in[5];
    const float* bc    = (const float*)d_in[6];
    const float* Wo    = (const float*)d_in[7];
    const float* bo    = (const float*)d_in[8];
    const float* Wout  = (const float*)d_in[9];
    const float* bout  = (const float*)d_in[10];

    float* ws = (float*)d_ws;
    const size_t HC = (size_t)BN * HN;            // 131072 floats
    float* hbuf[2] = { ws,          ws + HC     };
    float* cbuf[2] = { ws + 2 * HC, ws + 3 * HC };
    float* logits  = ws + 4 * HC;                 // [BN, OUTN]

    // zero h0 / c0 ping-pong buffers (workspace is poisoned by harness)
    const int nInit = (int)(4 * HC);
    lstm_init_kernel<<<(nInit + 255) / 256, 256, 0, stream>>>(ws, nInit);

    // 512 sequential steps: ping-pong h/c. SEQ even -> final state in buf 0.
    for (int t = 0; t < SEQN; ++t) {
        lstm_step_kernel<<<64, 256, 0, stream>>>(
            batch + (size_t)t * BN * INN,
            Wf, bf, Wi, bi, Wc, bc, Wo, bo,
            hbuf[t & 1], cbuf[t & 1],
            hbuf[(t + 1) & 1], cbuf[(t + 1) & 1]);
    }

    out_gemm_kernel<<<16, 256, 0, stream>>>(hbuf[0], Wout, bout, logits);
    logsoftmax_kernel<<<32, 256, 0, stream>>>(logits, (float*)d_out);
}